// GINEncoder_42820823941375
// MI455X (gfx1250) — compile-verified
//
#include <hip/hip_runtime.h>
#include <hip/hip_bf16.h>

typedef __attribute__((ext_vector_type(2))) float v2f;
typedef __attribute__((ext_vector_type(8))) float v8f;

#define EMB   300
#define P1    640   // padded hidden width (>=600, multiple of 64)
#define P2    320   // padded output width (>=300, multiple of 64)
#define NLAYERS 5
#define LDS_PAIR_STRIDE 160   // floats per K-pair row (80 float2; 80 % 32 == 16 -> conflict-free halves)

// ---------------- edge scatter: neigh[dst] += h[src] ----------------
__global__ __launch_bounds__(256) void edge_scatter_kernel(
    const float* __restrict__ h, const int* __restrict__ src,
    const int* __restrict__ dst, float* __restrict__ neigh, int E)
{
    int t = blockIdx.x * blockDim.x + threadIdx.x;
    if (t >= E * 75) return;               // 75 float4 chunks per 300-float row
    int e = t / 75, q = t % 75;
    int s = src[e], d = dst[e];
    const float4 v = *(const float4*)(h + (size_t)s * EMB + 4 * q);
    float* p = neigh + (size_t)d * EMB + 4 * q;
    atomicAdd(p + 0, v.x);
    atomicAdd(p + 1, v.y);
    atomicAdd(p + 2, v.z);
    atomicAdd(p + 3, v.w);
}

// ---------------- x = (1+eps[l])*h + neigh (may write in place) ----------------
__global__ __launch_bounds__(256) void combine_kernel(
    const float* __restrict__ h, const float* __restrict__ neigh,
    const float* __restrict__ epsArr, int l, float* __restrict__ x, int n)
{
    int i = blockIdx.x * blockDim.x + threadIdx.x;
    if (i >= n) return;
    float e = 1.0f + epsArr[l];
    x[i] = e * h[i] + neigh[i];
}

// ---------------- weight / bias padding ----------------
__global__ __launch_bounds__(256) void pad_w_kernel(
    const float* __restrict__ W, float* __restrict__ Wp,
    int rows, int cols, int pcols)
{
    int i = blockIdx.x * blockDim.x + threadIdx.x;
    if (i >= rows * pcols) return;
    int r = i / pcols, c = i % pcols;
    Wp[i] = (c < cols) ? W[(size_t)r * cols + c] : 0.0f;
}

__global__ __launch_bounds__(256) void pad_b_kernel(
    const float* __restrict__ b, float* __restrict__ bp, int cols, int pcols)
{
    int i = blockIdx.x * blockDim.x + threadIdx.x;
    if (i >= pcols) return;
    bp[i] = (i < cols) ? b[i] : 0.0f;
}

// ---------------- WMMA fp32 GEMM with LDS-staged weights ----------------
// C[nrows x ldc] = A[nrows x lda(K)] * B[K x ldc] + bias, optional ReLU.
// Block: 256 threads = 8 waves; each wave computes a 16x64 tile (4 accumulators).
// B panel (Krun x 64) is staged in LDS as interleaved (k, k+1) pairs so each
// lane's B fragment is one ds_load_b64.
extern "C" __global__ __launch_bounds__(256) void gemm_bias_kernel(
    const float* __restrict__ A, int lda, int nrows, int Krun,
    const float* __restrict__ B, const float* __restrict__ bias,
    float* __restrict__ C, int ldc, int do_relu)
{
    extern __shared__ float ldsB[];

    const int n0 = blockIdx.x * 64;

    // ---- cooperative LDS fill of the 64-wide B panel ----
    {
        const int total = Krun * 64;
        for (int idx = threadIdx.x; idx < total; idx += 256) {
            int k = idx >> 6;
            int c = idx & 63;
            ldsB[(k >> 1) * LDS_PAIR_STRIDE + (c << 1) + (k & 1)] =
                B[(size_t)k * ldc + n0 + c];
        }
    }
    __syncthreads();

    const int lane = threadIdx.x & 31;
    const int wave = threadIdx.x >> 5;
    const int rowTile = blockIdx.y * 8 + wave;
    if (rowTile * 16 < nrows) {            // wave-uniform guard; EXEC all-1s inside
        const int m0   = rowTile * 16;
        const int csel = lane & 15;        // col within 16-tile / A row select
        const int ksel = (lane >> 4) << 1; // K sub-offset: 0 or 2
        const int rowA = m0 + csel;

        v8f acc[4] = {};

        const float* Ap = A + (size_t)rowA * lda + ksel;
        const float* Lbase = ldsB + (ksel >> 1) * LDS_PAIR_STRIDE + (csel << 1);

        for (int kg = 0; kg < Krun; kg += 20) {
            // prefetch 5 A fragments in one clause
            v2f a[5];
#pragma unroll
            for (int j = 0; j < 5; ++j)
                a[j] = *(const v2f*)(Ap + kg + 4 * j);

#pragma unroll
            for (int j = 0; j < 5; ++j) {
                const float* Lp = Lbase + ((kg + 4 * j) >> 1) * LDS_PAIR_STRIDE;
#pragma unroll
                for (int t = 0; t < 4; ++t) {
                    v2f b = *(const v2f*)(Lp + 32 * t);   // {B[kb][col], B[kb+1][col]}
                    acc[t] = __builtin_amdgcn_wmma_f32_16x16x4_f32(
                        false, a[j], false, b, (short)0, acc[t], false, false);
                }
            }
        }

        const int rhi = (lane >> 4) << 3;  // C rows: vgpr r -> m = r (+8 for upper lanes)
#pragma unroll
        for (int t = 0; t < 4; ++t) {
            const int col = n0 + 16 * t + csel;
            const float bb = bias[col];
            float* Cp = C + (size_t)(m0 + rhi) * ldc + col;
#pragma unroll
            for (int r = 0; r < 8; ++r) {
                float v = acc[t][r] + bb;
                if (do_relu) v = fmaxf(v, 0.0f);
                Cp[(size_t)r * ldc] = v;
            }
        }
    }
}

// ---------------- batch-norm stats: per-column sum & sumsq ----------------
__global__ __launch_bounds__(320) void bn_stats_kernel(
    const float* __restrict__ z, int ldz, int N,
    float* __restrict__ sums, float* __restrict__ sqs)
{
    int c = threadIdx.x;
    if (c >= EMB) return;
    int r0 = blockIdx.x * 128;
    int r1 = r0 + 128; if (r1 > N) r1 = N;
    float s = 0.0f, q = 0.0f;
    for (int r = r0; r < r1; ++r) {
        float v = z[(size_t)r * ldz + c];
        s += v; q += v * v;
    }
    atomicAdd(&sums[c], s);
    atomicAdd(&sqs[c], q);
}

// ---------------- batch-norm apply (+optional ReLU) ----------------
__global__ __launch_bounds__(256) void bn_apply_kernel(
    const float* __restrict__ z, int ldz,
    const float* __restrict__ sums, const float* __restrict__ sqs,
    const float* __restrict__ gamma, const float* __restrict__ beta, int l,
    float* __restrict__ hout, int N, int do_relu)
{
    int i = blockIdx.x * blockDim.x + threadIdx.x;
    if (i >= N * EMB) return;
    int r = i / EMB, c = i % EMB;
    float invN = 1.0f / (float)N;
    float mu  = sums[c] * invN;
    float var = sqs[c] * invN - mu * mu;
    float g = gamma[l * EMB + c], be = beta[l * EMB + c];
    float v = (z[(size_t)r * ldz + c] - mu) * rsqrtf(var + 1e-5f) * g + be;
    if (do_relu) v = fmaxf(v, 0.0f);
    hout[i] = v;
}

// ---------------- per-graph readout ----------------
__global__ __launch_bounds__(256) void readout_scatter_kernel(
    const float* __restrict__ h, const int* __restrict__ gid,
    float* __restrict__ gsum, float* __restrict__ gcnt, int N)
{
    int t = blockIdx.x * blockDim.x + threadIdx.x;
    if (t >= N * 75) return;
    int n = t / 75, q = t % 75;
    int g = gid[n];
    const float4 v = *(const float4*)(h + (size_t)n * EMB + 4 * q);
    float* p = gsum + (size_t)g * EMB + 4 * q;
    atomicAdd(p + 0, v.x);
    atomicAdd(p + 1, v.y);
    atomicAdd(p + 2, v.z);
    atomicAdd(p + 3, v.w);
    if (q == 0) atomicAdd(&gcnt[g], 1.0f);
}

__global__ __launch_bounds__(256) void readout_div_kernel(
    const float* __restrict__ gsum, const float* __restrict__ gcnt,
    float* __restrict__ out, int G)
{
    int i = blockIdx.x * blockDim.x + threadIdx.x;
    if (i >= G * EMB) return;
    int g = i / EMB;
    out[i] = gsum[i] / fmaxf(gcnt[g], 1.0f);
}

extern "C" void kernel_launch(void* const* d_in, const int* in_sizes, int n_in,
                              void* d_out, int out_size, void* d_ws, size_t ws_size,
                              hipStream_t stream) {
    const float* node_feats = (const float*)d_in[0];
    const float* W1  = (const float*)d_in[1];
    const float* b1  = (const float*)d_in[2];
    const float* W2  = (const float*)d_in[3];
    const float* b2  = (const float*)d_in[4];
    const float* eps = (const float*)d_in[5];
    const float* gamma = (const float*)d_in[6];
    const float* beta  = (const float*)d_in[7];
    const int* src = (const int*)d_in[8];
    const int* dst = (const int*)d_in[9];
    const int* gid = (const int*)d_in[10];
    float* out = (float*)d_out;

    const int N = in_sizes[0] / EMB;
    const int E = in_sizes[8];
    const int G = out_size / EMB;

    // ---- workspace layout (floats; all offsets 16B aligned) ----
    float* ws   = (float*)d_ws;
    float* hbuf = ws;                       size_t off = (size_t)N * EMB;
    float* nbuf = ws + off;                 off += (size_t)N * P2;   // neigh (N*300) / z (N*320) union
    float* ybuf = ws + off;                 off += (size_t)N * P1;
    float* Wp1  = ws + off;                 off += (size_t)EMB * P1;
    float* Wp2  = ws + off;                 off += (size_t)(2 * EMB) * P2;
    float* b1p  = ws + off;                 off += P1;
    float* b2p  = ws + off;                 off += P2;
    float* sums = ws + off;                 off += P2;
    float* sqs  = ws + off;                 off += P2;
    float* gsum = ws + off;                 off += (size_t)G * EMB;
    float* gcnt = ws + off;                 off += (size_t)G;
    (void)ws_size; (void)n_in;

    const int rowTiles = (N + 15) / 16;
    const dim3 gemmBlk(256);
    const dim3 g1Grid(P1 / 64, (rowTiles + 7) / 8);
    const dim3 g2Grid(P2 / 64, (rowTiles + 7) / 8);
    const size_t sh1 = (size_t)(EMB / 2) * LDS_PAIR_STRIDE * sizeof(float);      // K=300 panel
    const size_t sh2 = (size_t)(2 * EMB / 2) * LDS_PAIR_STRIDE * sizeof(float);  // K=600 panel

    for (int l = 0; l < NLAYERS; ++l) {
        const float* hcur = (l == 0) ? node_feats : hbuf;

        // 1) neigh = segment_sum(h[src], dst)
        hipMemsetAsync(nbuf, 0, (size_t)N * EMB * sizeof(float), stream);
        {
            int tot = E * 75;
            edge_scatter_kernel<<<(tot + 255) / 256, 256, 0, stream>>>(hcur, src, dst, nbuf, E);
        }

        // 2) h = (1+eps)*h + neigh  -> hbuf
        {
            int tot = N * EMB;
            combine_kernel<<<(tot + 255) / 256, 256, 0, stream>>>(hcur, nbuf, eps, l, hbuf, tot);
        }

        // 3) pad layer weights/biases
        pad_w_kernel<<<(EMB * P1 + 255) / 256, 256, 0, stream>>>(
            W1 + (size_t)l * EMB * 2 * EMB, Wp1, EMB, 2 * EMB, P1);
        pad_w_kernel<<<(2 * EMB * P2 + 255) / 256, 256, 0, stream>>>(
            W2 + (size_t)l * 2 * EMB * EMB, Wp2, 2 * EMB, EMB, P2);
        pad_b_kernel<<<(P1 + 255) / 256, 256, 0, stream>>>(b1 + (size_t)l * 2 * EMB, b1p, 2 * EMB, P1);
        pad_b_kernel<<<(P2 + 255) / 256, 256, 0, stream>>>(b2 + (size_t)l * EMB, b2p, EMB, P2);

        // 4) y = relu(x @ W1 + b1)   [N x 640]
        gemm_bias_kernel<<<g1Grid, gemmBlk, sh1, stream>>>(
            hbuf, EMB, N, EMB, Wp1, b1p, ybuf, P1, 1);

        // 5) z = y @ W2 + b2         [N x 320] (aliases nbuf; neigh dead by now)
        gemm_bias_kernel<<<g2Grid, gemmBlk, sh2, stream>>>(
            ybuf, P1, N, 2 * EMB, Wp2, b2p, nbuf, P2, 0);

        // 6) batch norm (+ReLU except last layer) -> hbuf
        hipMemsetAsync(sums, 0, 2 * P2 * sizeof(float), stream);   // sums & sqs contiguous
        bn_stats_kernel<<<(N + 127) / 128, 320, 0, stream>>>(nbuf, P2, N, sums, sqs);
        {
            int tot = N * EMB;
            bn_apply_kernel<<<(tot + 255) / 256, 256, 0, stream>>>(
                nbuf, P2, sums, sqs, gamma, beta, l, hbuf, N, (l < NLAYERS - 1) ? 1 : 0);
        }
    }

    // ---- per-graph mean readout ----
    hipMemsetAsync(gsum, 0, ((size_t)G * EMB + G) * sizeof(float), stream);  // gsum & gcnt contiguous
    {
        int tot = N * 75;
        readout_scatter_kernel<<<(tot + 255) / 256, 256, 0, stream>>>(hbuf, gid, gsum, gcnt, N);
    }
    readout_div_kernel<<<(G * EMB + 255) / 256, 256, 0, stream>>>(gsum, gcnt, out, G);
}